// DkNN_layer_26620207301313
// MI455X (gfx1250) — compile-verified
//
#include <hip/hip_runtime.h>

// Problem constants (fixed by the reference).
constexpr int Bsz = 8192;      // batch
constexpr int K   = 8;         // layers
constexpr int L   = 1000;      // labels
constexpr int Cn  = 100000;    // calibration size
constexpr int NPAD = 131072;   // next pow2 >= Cn, padded with +INF for bitonic sort
constexpr int BUCKET = 32;     // elements per router bucket (Cn % BUCKET == 0)
constexpr int ROUTER_N = Cn / BUCKET;  // 3125

typedef __attribute__((ext_vector_type(2))) float v2f;
typedef __attribute__((ext_vector_type(8))) float v8f;

// ---------------------------------------------------------------------------
// Sort pipeline: pad -> bitonic (LDS-fused local stages + global merges)
// ---------------------------------------------------------------------------
__global__ __launch_bounds__(256) void pad_kernel(const float* __restrict__ cali,
                                                  float* __restrict__ dst) {
  int i = blockIdx.x * 256 + threadIdx.x;
  if (i < NPAD) dst[i] = (i < Cn) ? cali[i] : __builtin_inff();
}

// Full bitonic sort of 2048-element blocks in LDS (covers k = 2..2048).
__global__ __launch_bounds__(256) void bitonic_local_sort(float* __restrict__ data) {
  __shared__ float s[2048];
  const int t = threadIdx.x;
  const int base = blockIdx.x * 2048;
  for (int i = t; i < 2048; i += 256) s[i] = data[base + i];
  __syncthreads();
  for (int k = 2; k <= 2048; k <<= 1) {
    for (int j = k >> 1; j >= 1; j >>= 1) {
      for (int p = t; p < 1024; p += 256) {
        int i = ((p & ~(j - 1)) << 1) | (p & (j - 1));
        int m = i + j;
        bool up = (((base + i) & k) == 0);
        float a = s[i], c = s[m];
        if ((a > c) == up) { s[i] = c; s[m] = a; }
      }
      __syncthreads();
    }
  }
  for (int i = t; i < 2048; i += 256) data[base + i] = s[i];
}

// One global compare-exchange stage (stride j >= 2048).
__global__ __launch_bounds__(256) void bitonic_global_merge(float* __restrict__ data,
                                                            int j, int k) {
  int p = blockIdx.x * 256 + threadIdx.x;            // pair index, < NPAD/2
  int i = ((p & ~(j - 1)) << 1) | (p & (j - 1));
  int m = i + j;
  bool up = ((i & k) == 0);
  float a = data[i], c = data[m];
  if ((a > c) == up) { data[i] = c; data[m] = a; }
}

// Finish strides j = 1024..1 of outer stage k (k >= 4096) inside LDS.
__global__ __launch_bounds__(256) void bitonic_local_merge(float* __restrict__ data,
                                                           int k) {
  __shared__ float s[2048];
  const int t = threadIdx.x;
  const int base = blockIdx.x * 2048;
  for (int i = t; i < 2048; i += 256) s[i] = data[base + i];
  __syncthreads();
  const bool up = ((base & k) == 0);   // k >= 4096: direction constant per block
  for (int j = 1024; j >= 1; j >>= 1) {
    for (int p = t; p < 1024; p += 256) {
      int i = ((p & ~(j - 1)) << 1) | (p & (j - 1));
      int m = i + j;
      float a = s[i], c = s[m];
      if ((a > c) == up) { s[i] = c; s[m] = a; }
    }
    __syncthreads();
  }
  for (int i = t; i < 2048; i += 256) data[base + i] = s[i];
}

__global__ __launch_bounds__(256) void router_extract(const float* __restrict__ sorted,
                                                      float* __restrict__ router) {
  int i = blockIdx.x * 256 + threadIdx.x;
  if (i < ROUTER_N) router[i] = sorted[i * BUCKET];
}

// ---------------------------------------------------------------------------
// Main kernel: WMMA K-reduction + two-level rank search.
//   Level 1: fully-unrolled branchless lower_bound on a 12.5 KB LDS router.
//   Level 2: fetch the whole 128 B bucket (8 independent b128 loads, one wait)
//            and count elements < s with VALU compares.
// Grid: (4, Bsz/32), block 256 (8 waves). Wave "slot" = blockIdx.x*8 + waveId
// covers l in [slot*32, slot*32+32). Each block loops over 32 consecutive b's
// to amortize the router load. Streaming loads/stores are non-temporal so the
// 262 MB input stream does not evict the L2-resident sorted table.
// ---------------------------------------------------------------------------
__global__ __launch_bounds__(256) void pvalue_kernel(const float* __restrict__ A,
                                                     const float* __restrict__ sorted,
                                                     const float* __restrict__ router,
                                                     float* __restrict__ out) {
  __shared__ float r[ROUTER_N];
  for (int i = threadIdx.x; i < ROUTER_N; i += 256) r[i] = router[i];
  __syncthreads();

  const int lane = threadIdx.x & 31;
  const int wave = threadIdx.x >> 5;
  const int slot = blockIdx.x * 8 + wave;   // 0..31
  const int col  = lane & 15;
  const int half = lane >> 4;               // 0: rows K0/K1 ; 1: rows K2/K3
  const int k0   = half << 1;               // 0 or 2
  const int lA   = slot * 32 + col;         // tile A column for this lane
  const int lB   = lA + 16;                 // tile B column
  const int lAc  = (lA < L) ? lA : (L - 1); // branchless clamp: keep EXEC all-ones
  const int lBc  = (lB < L) ? lB : (L - 1);
  const int l    = slot * 32 + lane;        // this lane's output label

  v2f ones; ones.x = 1.0f; ones.y = 1.0f;

  for (int bb = 0; bb < 32; ++bb) {
    const int b = (blockIdx.y << 5) + bb;
    const float* base = A + (size_t)b * (K * L);

    // B-matrix (4x16 f32): VGPR0 = rows {K0 (lanes 0-15), K2 (lanes 16-31)},
    // VGPR1 = rows {K1, K3}. Two tiles (lA, lB); second WMMA pair covers k=4..7.
    // Non-temporal: each element is read exactly once.
    v2f bA0, bA1, bB0, bB1;
    bA0.x = __builtin_nontemporal_load(&base[(k0 + 0) * L + lAc]);
    bA0.y = __builtin_nontemporal_load(&base[(k0 + 1) * L + lAc]);
    bA1.x = __builtin_nontemporal_load(&base[(k0 + 4) * L + lAc]);
    bA1.y = __builtin_nontemporal_load(&base[(k0 + 5) * L + lAc]);
    bB0.x = __builtin_nontemporal_load(&base[(k0 + 0) * L + lBc]);
    bB0.y = __builtin_nontemporal_load(&base[(k0 + 1) * L + lBc]);
    bB1.x = __builtin_nontemporal_load(&base[(k0 + 4) * L + lBc]);
    bB1.y = __builtin_nontemporal_load(&base[(k0 + 5) * L + lBc]);

    // Prefetch next b's cachelines (128B lines cover both lA and lB columns).
    if (bb < 31) {
      const float* nb = base + K * L;
      __builtin_prefetch(nb + (k0 + 0) * L + lAc, 0, 1);
      __builtin_prefetch(nb + (k0 + 1) * L + lAc, 0, 1);
      __builtin_prefetch(nb + (k0 + 4) * L + lAc, 0, 1);
      __builtin_prefetch(nb + (k0 + 5) * L + lAc, 0, 1);
    }

    // D = ones(16x4) x B(4x16) + C  =>  every row of D = column sums of B.
    v8f cA = {};
    cA = __builtin_amdgcn_wmma_f32_16x16x4_f32(false, ones, false, bA0,
                                               (short)0, cA, false, false);
    cA = __builtin_amdgcn_wmma_f32_16x16x4_f32(false, ones, false, bA1,
                                               (short)0, cA, false, false);
    v8f cB = {};
    cB = __builtin_amdgcn_wmma_f32_16x16x4_f32(false, ones, false, bB0,
                                               (short)0, cB, false, false);
    cB = __builtin_amdgcn_wmma_f32_16x16x4_f32(false, ones, false, bB1,
                                               (short)0, cB, false, false);

    // D VGPR0: lanes 0-15 hold (M=0, N=lane), lanes 16-31 hold (M=8, N=lane-16);
    // rows are replicated, so every lane's c[0] is total[b, slot*32 + lane].
    float s = (half == 0) ? cA[0] : cB[0];

    // Level 1: fully-unrolled branchless lower_bound over LDS router.
    // Invariant: lo = confirmed count of router entries < s.
    int lo = 0;
#pragma unroll
    for (int step = 2048; step >= 1; step >>= 1) {
      int cand = lo + step;
      int inb  = (cand <= ROUTER_N);               // 0/1, no short-circuit branch
      int idx  = inb ? (cand - 1) : (ROUTER_N - 1);  // clamp: LDS read always valid
      int ok   = inb & (r[idx] < s);
      lo = ok ? cand : lo;
    }

    // Level 2: load the whole 32-element (128 B, 128 B-aligned) bucket with
    // 8 independent b128 loads, then count elements < s. Branchless.
    int bkt = ((lo > 0) ? (lo - 1) : 0) * BUCKET;
    const float4* bp = (const float4*)(sorted + bkt);
    int cnt = 0;
#pragma unroll
    for (int q = 0; q < 8; ++q) {
      float4 v = bp[q];
      cnt += (v.x < s);
      cnt += (v.y < s);
      cnt += (v.z < s);
      cnt += (v.w < s);
    }
    int pos = (lo == 0) ? 0 : (bkt + cnt);

    float p = (float)(Cn - pos) * (1.0f / (float)Cn);
    if (l < L)
      __builtin_nontemporal_store(p, &out[(size_t)b * L + l]);  // 32 consecutive floats/wave
  }
}

// ---------------------------------------------------------------------------
extern "C" void kernel_launch(void* const* d_in, const int* in_sizes, int n_in,
                              void* d_out, int out_size, void* d_ws, size_t ws_size,
                              hipStream_t stream) {
  (void)in_sizes; (void)n_in; (void)out_size; (void)ws_size;
  const float* noncf = (const float*)d_in[0];   // [B,K,L] f32
  const float* cali  = (const float*)d_in[2];   // [C] f32   (d_in[1] = label ids, unused)
  float* out    = (float*)d_out;
  float* sorted = (float*)d_ws;                 // NPAD floats (512 KB)
  float* router = sorted + NPAD;                // ROUTER_N floats (12.5 KB)

  pad_kernel<<<NPAD / 256, 256, 0, stream>>>(cali, sorted);
  bitonic_local_sort<<<NPAD / 2048, 256, 0, stream>>>(sorted);
  for (int k = 4096; k <= NPAD; k <<= 1) {
    for (int j = k >> 1; j >= 2048; j >>= 1)
      bitonic_global_merge<<<NPAD / 512, 256, 0, stream>>>(sorted, j, k);
    bitonic_local_merge<<<NPAD / 2048, 256, 0, stream>>>(sorted, k);
  }
  router_extract<<<(ROUTER_N + 255) / 256, 256, 0, stream>>>(sorted, router);

  pvalue_kernel<<<dim3(4, Bsz / 32), 256, 0, stream>>>(noncf, sorted, router, out);
}